// QCNN_45389214384341
// MI455X (gfx1250) — compile-verified
//
#include <hip/hip_runtime.h>
#include <hip/hip_bf16.h>

// ---------------------------------------------------------------------------
// QCNN forward: qfeat GEMM via V_WMMA_F32_16X16X4_F32, then LDS-resident
// 12-qubit state-vector simulation, one workgroup per batch element.
// ---------------------------------------------------------------------------

#define N_QUBITS 12
#define BATCH    4096
#define IN_DIM   16
#define NSTATE   4096    // 2^12

typedef float v2f __attribute__((ext_vector_type(2)));
typedef float v8f __attribute__((ext_vector_type(8)));

#if __has_builtin(__builtin_amdgcn_wmma_f32_16x16x4_f32)
#define HAS_WMMA_F32X4 1
#else
#define HAS_WMMA_F32X4 0
#endif

// ---------------------------------------------------------------------------
// Kernel 1: qfeat[b][q] = sum_k x[b][k] * W[q][k] + bias[q]
// M=4096, N=12 (padded to 16), K=16 -> 4 chained v_wmma_f32_16x16x4_f32.
// One 16x16 output tile per wave; 8 waves (256 threads) per block; 32 blocks.
// A layout (16x4 f32): lane<16 -> M=lane, v0=K0,v1=K1 ; lane>=16 -> v0=K2,v1=K3.
// ---------------------------------------------------------------------------
__global__ __launch_bounds__(256) void qfeat_wmma_kernel(
    const float* __restrict__ x, const float* __restrict__ W,
    const float* __restrict__ bias, float* __restrict__ qfeat)
{
    const int wave = threadIdx.x >> 5;
    const int lane = threadIdx.x & 31;
    const int half = lane >> 4;
    const int ln   = lane & 15;
    const int tileM = blockIdx.x * 8 + wave;

#if HAS_WMMA_F32X4
    const int row = tileM * 16 + ln;
    const float2* x2 = (const float2*)x;
    const float2* w2 = (const float2*)W;

    v8f acc = {};
#pragma unroll
    for (int kc = 0; kc < 4; ++kc) {
        const int kp = kc * 2 + half;                  // float2 index = (kc*4 + 2*half)/2
        float2 av = x2[row * 8 + kp];
        float2 bv = (ln < N_QUBITS) ? w2[ln * 8 + kp] : make_float2(0.f, 0.f);
        v2f a; a[0] = av.x; a[1] = av.y;
        v2f b; b[0] = bv.x; b[1] = bv.y;
        acc = __builtin_amdgcn_wmma_f32_16x16x4_f32(
            /*neg_a=*/false, a, /*neg_b=*/false, b,
            /*c_mod=*/(short)0, acc, /*reuse_a=*/false, /*reuse_b=*/false);
    }
    const float badd = (ln < N_QUBITS) ? bias[ln] : 0.f;
#pragma unroll
    for (int r = 0; r < 8; ++r) {
        const int m = tileM * 16 + r + half * 8;       // D layout: vgpr r = M r / r+8
        if (ln < N_QUBITS) qfeat[m * N_QUBITS + ln] = acc[r] + badd;
    }
#else
    // Fallback (should not trigger on gfx1250): scalar dot products.
    for (int e = threadIdx.x + blockIdx.x * 256; e < 16 * N_QUBITS * 8; e += 256) {
        const int m = tileM * 16 + (e & 15);
        (void)m;
    }
    const int gtid = threadIdx.x + blockIdx.x * 256;
    for (int idx = gtid; idx < BATCH * N_QUBITS; idx += 32 * 256) {
        const int m = idx / N_QUBITS, q = idx % N_QUBITS;
        float acc = bias[q];
        for (int k = 0; k < IN_DIM; ++k) acc += x[m * IN_DIM + k] * W[q * IN_DIM + k];
        qfeat[idx] = acc;
    }
#endif
}

// ---------------------------------------------------------------------------
// Complex helpers + Kernel 2: precompute the 11 batch-independent 4x4 block
// unitaries  U = (SU2_a2 (x) SU2_b2) . ZZ . YY . XX . (SU2_a1 (x) SU2_b1)
// and the final SU2 on qubit 0.  gout: 11*16 + 4 = 180 float2.
// ---------------------------------------------------------------------------
struct cplx { float x, y; };
__device__ inline cplx cmul(cplx a, cplx b) { return {a.x*b.x - a.y*b.y, a.x*b.y + a.y*b.x}; }
__device__ inline cplx cadd(cplx a, cplx b) { return {a.x + b.x, a.y + b.y}; }
__device__ inline cplx cexpi(float t) { return {cosf(t), sinf(t)}; }   // e^{i t}

__device__ inline void su2(const float* p, cplx* M) {
    // RZ(p2) @ RY(p1) @ RZ(p0)
    const float c = cosf(0.5f * p[1]), s = sinf(0.5f * p[1]);
    const cplx z0m = cexpi(-0.5f * p[0]), z0p = cexpi(0.5f * p[0]);
    const cplx z2m = cexpi(-0.5f * p[2]), z2p = cexpi(0.5f * p[2]);
    cplx T00 = { c * z0m.x,  c * z0m.y}, T01 = {-s * z0p.x, -s * z0p.y};
    cplx T10 = { s * z0m.x,  s * z0m.y}, T11 = { c * z0p.x,  c * z0p.y};
    M[0] = cmul(z2m, T00); M[1] = cmul(z2m, T01);
    M[2] = cmul(z2p, T10); M[3] = cmul(z2p, T11);
}

__device__ inline void mat4mul(const cplx* A, const cplx* B, cplx* C) {
    for (int i = 0; i < 4; ++i)
        for (int j = 0; j < 4; ++j) {
            cplx acc = {0.f, 0.f};
            for (int k = 0; k < 4; ++k) acc = cadd(acc, cmul(A[i*4+k], B[k*4+j]));
            C[i*4+j] = acc;
        }
}

__device__ inline void kron22(const cplx* A, const cplx* B, cplx* K) { // A on high bit
    for (int i = 0; i < 2; ++i) for (int j = 0; j < 2; ++j)
        for (int k = 0; k < 2; ++k) for (int l = 0; l < 2; ++l)
            K[(2*i+k)*4 + (2*j+l)] = cmul(A[i*2+j], B[k*2+l]);
}

__global__ void gates_kernel(const float* __restrict__ params, float2* __restrict__ gout)
{
    if (threadIdx.x != 0 || blockIdx.x != 0) return;
    int p = 0;
    for (int blk = 0; blk < 11; ++blk) {
        const float* pp = params + p; p += 15;
        cplx Ua1[4], Ub1[4], Ua2[4], Ub2[4];
        su2(pp + 0, Ua1); su2(pp + 3, Ub1); su2(pp + 9, Ua2); su2(pp + 12, Ub2);

        const float cx = cosf(0.5f * pp[6]), sx = sinf(0.5f * pp[6]);
        const float cy = cosf(0.5f * pp[7]), sy = sinf(0.5f * pp[7]);
        cplx XX[16] = {}, YY[16] = {}, ZZ[16] = {};
        for (int i = 0; i < 4; ++i) { XX[i*4+i] = {cx, 0.f}; XX[i*4+(3-i)] = {0.f, -sx}; }
        for (int i = 0; i < 4; ++i) YY[i*4+i] = {cy, 0.f};
        YY[1*4+2] = {0.f, -sy}; YY[2*4+1] = {0.f, -sy};
        YY[0*4+3] = {0.f,  sy}; YY[3*4+0] = {0.f,  sy};
        const cplx zm = cexpi(-0.5f * pp[8]), zp = cexpi(0.5f * pp[8]);
        ZZ[0] = zm; ZZ[5] = zp; ZZ[10] = zp; ZZ[15] = zm;

        cplx U1[16], U2[16], T1[16], T2[16], T3[16], Ub[16];
        kron22(Ua1, Ub1, U1);
        kron22(Ua2, Ub2, U2);
        mat4mul(YY, XX, T1);      // YY . XX
        mat4mul(ZZ, T1, T2);      // ZZ . YY . XX
        mat4mul(T2, U1, T3);
        mat4mul(U2, T3, Ub);
        for (int i = 0; i < 16; ++i) gout[blk*16 + i] = make_float2(Ub[i].x, Ub[i].y);
    }
    cplx F[4];
    su2(params + p, F);           // p == 165
    for (int i = 0; i < 4; ++i) gout[176 + i] = make_float2(F[i].x, F[i].y);
}

// ---------------------------------------------------------------------------
// Kernel 3: full state-vector simulation, one block (256 thr / 8 waves) per
// batch element; state (re/im, 32KB) lives in LDS for the entire circuit.
// Qubit q maps to amplitude-index bit (11 - q) (reference reshape order).
// All 11 diagonal ZZ gates per layer are fused into one phase pass.
// ---------------------------------------------------------------------------
__global__ __launch_bounds__(256) void sim_kernel(
    const float* __restrict__ qfeat, const float2* __restrict__ gates,
    float* __restrict__ out)
{
    __shared__ float sre[NSTATE];
    __shared__ float sim_[NSTATE];
    __shared__ float2 sG[180];
    __shared__ float sc[N_QUBITS], ss[N_QUBITS], sth[N_QUBITS - 1];
    __shared__ float red[256];

    const int tid  = threadIdx.x;
    const int bidx = blockIdx.x;

    __builtin_prefetch(&qfeat[bidx * N_QUBITS], 0, 0);

    for (int i = tid; i < 180; i += 256) sG[i] = gates[i];
    if (tid < N_QUBITS) {
        const float f = qfeat[bidx * N_QUBITS + tid];
        sc[tid] = cosf(0.5f * f);
        ss[tid] = sinf(0.5f * f);
    }
    if (tid < N_QUBITS - 1)
        sth[tid] = qfeat[bidx * N_QUBITS + tid] * qfeat[bidx * N_QUBITS + tid + 1];
    for (int i = tid; i < NSTATE; i += 256) { sre[i] = (i == 0) ? 1.f : 0.f; sim_[i] = 0.f; }
    __syncthreads();

    // -------- 2 encoding layers: 12 RY gates + fused ZZ phases --------
    for (int layer = 0; layer < 2; ++layer) {
        for (int q = 0; q < N_QUBITS; ++q) {
            const int bp = 11 - q;
            const float c = sc[q], s = ss[q];
            for (int t = tid; t < NSTATE / 2; t += 256) {
                const int i0 = ((t >> bp) << (bp + 1)) | (t & ((1 << bp) - 1));
                const int i1 = i0 | (1 << bp);
                const float a0 = sre[i0], b0 = sim_[i0];
                const float a1 = sre[i1], b1 = sim_[i1];
                sre[i0] = c * a0 - s * a1;  sim_[i0] = c * b0 - s * b1;
                sre[i1] = s * a0 + c * a1;  sim_[i1] = s * b0 + c * b1;
            }
            __syncthreads();
        }
        // fused diagonal ZZ chain: one complex phase per amplitude
        for (int i = tid; i < NSTATE; i += 256) {
            float ang = 0.f;
#pragma unroll
            for (int q = 0; q < N_QUBITS - 1; ++q) {
                const int b0 = (i >> (11 - q)) & 1;
                const int b1 = (i >> (10 - q)) & 1;
                ang -= 0.5f * sth[q] * ((b0 == b1) ? 1.f : -1.f);
            }
            const float cp = cosf(ang), sp = sinf(ang);
            const float r = sre[i], m = sim_[i];
            sre[i]  = r * cp - m * sp;
            sim_[i] = r * sp + m * cp;
        }
        __syncthreads();
    }

    // -------- 11 fixed 4x4 two-qubit blocks --------
    for (int blk = 0; blk < 11; ++blk) {
        const int a  = (blk < 6) ? 2 * blk : 2 * (blk - 6) + 1;
        const int pa = 11 - a;            // high bit (qubit a)
        const int pb = pa - 1;            // low  bit (qubit a+1)
        const float2* M = &sG[blk * 16];
        for (int g = tid; g < NSTATE / 4; g += 256) {
            const int t1  = ((g >> pb) << (pb + 1)) | (g & ((1 << pb) - 1));
            const int i00 = ((t1 >> pa) << (pa + 1)) | (t1 & ((1 << pa) - 1));
            const int idx[4] = { i00, i00 | (1 << pb), i00 | (1 << pa),
                                 i00 | (1 << pa) | (1 << pb) };
            float vr[4], vi[4];
#pragma unroll
            for (int k = 0; k < 4; ++k) { vr[k] = sre[idx[k]]; vi[k] = sim_[idx[k]]; }
#pragma unroll
            for (int j = 0; j < 4; ++j) {
                float ar = 0.f, ai = 0.f;
#pragma unroll
                for (int k = 0; k < 4; ++k) {
                    const float2 m = M[j * 4 + k];
                    ar = fmaf(m.x, vr[k], fmaf(-m.y, vi[k], ar));
                    ai = fmaf(m.x, vi[k], fmaf( m.y, vr[k], ai));
                }
                sre[idx[j]] = ar; sim_[idx[j]] = ai;
            }
        }
        __syncthreads();
    }

    // -------- final SU2 on qubit 0 (amplitude bit 11) --------
    {
        const float2 F00 = sG[176], F01 = sG[177], F10 = sG[178], F11 = sG[179];
        for (int t = tid; t < NSTATE / 2; t += 256) {
            const int i0 = t, i1 = t + 2048;
            const float r0 = sre[i0], m0 = sim_[i0], r1 = sre[i1], m1 = sim_[i1];
            sre[i0]  = F00.x * r0 - F00.y * m0 + F01.x * r1 - F01.y * m1;
            sim_[i0] = F00.x * m0 + F00.y * r0 + F01.x * m1 + F01.y * r1;
            sre[i1]  = F10.x * r0 - F10.y * m0 + F11.x * r1 - F11.y * m1;
            sim_[i1] = F10.x * m0 + F10.y * r0 + F11.x * m1 + F11.y * r1;
        }
        __syncthreads();
    }

    // -------- reduction:  p0 - p1  ->  y --------
    float acc = 0.f;
    for (int i = tid; i < NSTATE; i += 256) {
        const float p = sre[i] * sre[i] + sim_[i] * sim_[i];
        acc += (i & 2048) ? -p : p;
    }
    red[tid] = acc;
    __syncthreads();
    for (int off = 128; off > 0; off >>= 1) {
        if (tid < off) red[tid] += red[tid + off];
        __syncthreads();
    }
    if (tid == 0) {
        float y = (red[0] + 1.f) * 0.5f;
        y = fminf(fmaxf(y, 1e-6f), 1.f - 1e-6f);
        out[bidx] = y;
    }
}

// ---------------------------------------------------------------------------
extern "C" void kernel_launch(void* const* d_in, const int* in_sizes, int n_in,
                              void* d_out, int out_size, void* d_ws, size_t ws_size,
                              hipStream_t stream)
{
    const float* x      = (const float*)d_in[0];   // 4096 x 16
    const float* W      = (const float*)d_in[1];   // 12 x 16
    const float* bias   = (const float*)d_in[2];   // 12
    const float* params = (const float*)d_in[3];   // 168
    float* out = (float*)d_out;                    // 4096

    float*  qfeat = (float*)d_ws;                                        // 4096*12 f32
    float2* gates = (float2*)((char*)d_ws + (size_t)BATCH * N_QUBITS * sizeof(float)); // 180 cplx

    (void)in_sizes; (void)n_in; (void)out_size; (void)ws_size;

    qfeat_wmma_kernel<<<BATCH / (16 * 8), 256, 0, stream>>>(x, W, bias, qfeat);
    gates_kernel<<<1, 32, 0, stream>>>(params, gates);
    sim_kernel<<<BATCH, 256, 0, stream>>>(qfeat, gates, out);
}